// NEGLoss_13159779795490
// MI455X (gfx1250) — compile-verified
//
#include <hip/hip_runtime.h>
#include <hip/hip_bf16.h>
#include <math.h>

typedef __attribute__((ext_vector_type(2))) float v2f;
typedef __attribute__((ext_vector_type(8))) float v8f;

// Stable softplus: log1p(exp(t)) = max(t,0) + log1p(exp(-|t|))
__device__ __forceinline__ float softplus_f(float t) {
    return fmaxf(t, 0.0f) + log1pf(__expf(-fabsf(t)));
}

#define NBLK 1024   // number of partial sums (multiple of 64 for the WMMA reduce)
#define NTHR 256

// Lane-per-edge: each thread gathers 7 rows (src, pos, 5 neg) of x[*][128]
// as float4 streams and accumulates 6 dot products in f32 VALU FMAs.
__global__ void __launch_bounds__(NTHR)
neg_loss_partial_kernel(const float* __restrict__ x,
                        const int* __restrict__ edge_index,   // [2, E]
                        const int* __restrict__ neg_index,    // [K=5, E]
                        float* __restrict__ partial,          // [NBLK]
                        int E)
{
    const int tid    = blockIdx.x * blockDim.x + threadIdx.x;
    const int stride = gridDim.x * blockDim.x;
    const float4* __restrict__ X = reinterpret_cast<const float4*>(x); // 32 float4 per row

    float local = 0.0f;
    for (int e = tid; e < E; e += stride) {
        const float4* vs = X + (size_t)edge_index[e]         * 32;
        const float4* vp = X + (size_t)edge_index[E + e]     * 32;
        const float4* n0 = X + (size_t)neg_index[0 * E + e]  * 32;
        const float4* n1 = X + (size_t)neg_index[1 * E + e]  * 32;
        const float4* n2 = X + (size_t)neg_index[2 * E + e]  * 32;
        const float4* n3 = X + (size_t)neg_index[3 * E + e]  * 32;
        const float4* n4 = X + (size_t)neg_index[4 * E + e]  * 32;

        // Prefetch next grid-stride edge's rows (global_prefetch_b8) to hide
        // gather latency; rows are L2-resident (x = 51 MB << 192 MB L2).
        int en = e + stride;
        if (en < E) {
            __builtin_prefetch(X + (size_t)edge_index[en]        * 32, 0, 0);
            __builtin_prefetch(X + (size_t)edge_index[E + en]    * 32, 0, 0);
            __builtin_prefetch(X + (size_t)neg_index[0 * E + en] * 32, 0, 0);
            __builtin_prefetch(X + (size_t)neg_index[1 * E + en] * 32, 0, 0);
            __builtin_prefetch(X + (size_t)neg_index[2 * E + en] * 32, 0, 0);
            __builtin_prefetch(X + (size_t)neg_index[3 * E + en] * 32, 0, 0);
            __builtin_prefetch(X + (size_t)neg_index[4 * E + en] * 32, 0, 0);
        }

        float dp = 0.f, d0 = 0.f, d1 = 0.f, d2 = 0.f, d3 = 0.f, d4 = 0.f;
        #pragma unroll 4
        for (int i = 0; i < 32; ++i) {
            float4 v = vs[i];
            float4 t;
            t = vp[i]; dp = fmaf(v.x, t.x, fmaf(v.y, t.y, fmaf(v.z, t.z, fmaf(v.w, t.w, dp))));
            t = n0[i]; d0 = fmaf(v.x, t.x, fmaf(v.y, t.y, fmaf(v.z, t.z, fmaf(v.w, t.w, d0))));
            t = n1[i]; d1 = fmaf(v.x, t.x, fmaf(v.y, t.y, fmaf(v.z, t.z, fmaf(v.w, t.w, d1))));
            t = n2[i]; d2 = fmaf(v.x, t.x, fmaf(v.y, t.y, fmaf(v.z, t.z, fmaf(v.w, t.w, d2))));
            t = n3[i]; d3 = fmaf(v.x, t.x, fmaf(v.y, t.y, fmaf(v.z, t.z, fmaf(v.w, t.w, d3))));
            t = n4[i]; d4 = fmaf(v.x, t.x, fmaf(v.y, t.y, fmaf(v.z, t.z, fmaf(v.w, t.w, d4))));
        }

        // BCEWithLogits: positive contributes softplus(z)-z = softplus(-z);
        // negatives contribute softplus(z).
        local += softplus_f(-dp)
               + softplus_f(d0) + softplus_f(d1) + softplus_f(d2)
               + softplus_f(d3) + softplus_f(d4);
    }

    __shared__ float sdata[NTHR];
    sdata[threadIdx.x] = local;
    __syncthreads();
    #pragma unroll
    for (int s = NTHR / 2; s > 0; s >>= 1) {
        if ((int)threadIdx.x < s) sdata[threadIdx.x] += sdata[threadIdx.x + s];
        __syncthreads();
    }
    if (threadIdx.x == 0) partial[blockIdx.x] = sdata[0];
}

// Single-wave finalize. Each lane folds NBLK partials into two f32
// accumulators -> a full 16x4 f32 A-matrix (2 VGPRs x 32 lanes = 64 slots).
// One V_WMMA_F32_16X16X4_F32 against a ones B-matrix produces per-row sums
// (broadcast across columns); summing the 8 C/D VGPRs per lane and adding
// lane0 (rows 0-7) + lane16 (rows 8-15) gives the exact f32 total.
__global__ void neg_loss_finalize_kernel(const float* __restrict__ partial,
                                         float* __restrict__ out,
                                         int P, float invCount)
{
    const int l = threadIdx.x;   // 32 threads, EXEC all ones (WMMA requirement)
    float s0 = 0.f, s1 = 0.f;
    for (int i = l; i < P; i += 64)      s0 += partial[i];
    for (int i = l + 32; i < P; i += 64) s1 += partial[i];

    v2f a; a.x = s0;  a.y = s1;
    v2f b; b.x = 1.f; b.y = 1.f;
    v8f c = {};
    v8f d = __builtin_amdgcn_wmma_f32_16x16x4_f32(
        /*neg_a=*/false, a, /*neg_b=*/false, b,
        /*c_mod=*/(short)0, c, /*reuse_a=*/false, /*reuse_b=*/false);

    float t = d[0] + d[1] + d[2] + d[3] + d[4] + d[5] + d[6] + d[7];
    float t_hi = __shfl(t, 16, 32);   // lanes 16-31 hold rows 8..15
    if (l == 0) out[0] = (t + t_hi) * invCount;
}

extern "C" void kernel_launch(void* const* d_in, const int* in_sizes, int n_in,
                              void* d_out, int out_size, void* d_ws, size_t ws_size,
                              hipStream_t stream) {
    const float* x  = (const float*)d_in[0];   // [N, 128] f32
    const int*   ei = (const int*)d_in[1];     // [2, E]
    const int*   ni = (const int*)d_in[2];     // [5, E]
    const int E = in_sizes[1] / 2;
    const int K = (E > 0) ? (in_sizes[2] / E) : 5;   // = 5

    float* partial = (float*)d_ws;             // NBLK floats of scratch

    neg_loss_partial_kernel<<<NBLK, NTHR, 0, stream>>>(x, ei, ni, partial, E);

    const float invCount = 1.0f / ((float)E * (float)(K + 1));
    neg_loss_finalize_kernel<<<1, 32, 0, stream>>>(partial, (float*)d_out, NBLK, invCount);
}